// RegionProposalNetwork_60258391163291
// MI455X (gfx1250) — compile-verified
//
#include <hip/hip_runtime.h>
#include <hip/hip_bf16.h>

typedef __attribute__((ext_vector_type(2))) float v2f;
typedef __attribute__((ext_vector_type(8))) float v8f;

#define BATCH 8
#define CCH   512
#define HH    64
#define WW    64
#define ANC   9
#define HWSZ  (HH * WW)                 // 4096
#define MTOT  (BATCH * HWSZ)            // 32768
#define KTOT  (CCH * 9)                 // 4608
#define NHEAD 48                        // 36 bbox + 9 cls + 3 pad
#define PRE_NMS  2000
#define POST_NMS 1000
#define NMS_T    0.7f

#define KCH   16                        // channels staged per pipeline stage
#define NSTG  (9 * (CCH / KCH))         // 288 stages
#define LAPITCH 17                      // lA row pitch (pad: stride 17 coprime with 64 banks)

// ---------------------------------------------------------------------------
// Async global->LDS copy (CDNA5 GLOBAL_LOAD_ASYNC_TO_LDS_B32, ASYNCcnt).
// Guarded: falls back to a synchronous copy if the builtins are not declared.
// Builtin signature (from hipcc diagnostic): param0 = AS(1) int* (global src),
// param1 = AS(3) int* (LDS dst), then imm offset, imm cpol.
// ---------------------------------------------------------------------------
#if defined(__HIP_DEVICE_COMPILE__)
#if __has_builtin(__builtin_amdgcn_global_load_async_to_lds_b32) && \
    __has_builtin(__builtin_amdgcn_s_wait_asynccnt)
#define USE_ASYNC_LDS 1
#endif
#endif

#if defined(USE_ASYNC_LDS)
typedef __attribute__((address_space(1))) int gas_int;
typedef __attribute__((address_space(3))) int las_int;
__device__ __forceinline__ void async_copy_b32(const float* gsrc, float* ldst) {
    __builtin_amdgcn_global_load_async_to_lds_b32((gas_int*)gsrc, (las_int*)ldst, 0, 0);
}
#define WAIT_ASYNC_LE(n) __builtin_amdgcn_s_wait_asynccnt(n)
#else
__device__ __forceinline__ void async_copy_b32(const float* gsrc, float* ldst) {
    *ldst = *gsrc;
}
#define WAIT_ASYNC_LE(n) ((void)0)
#endif

// ---------------------------------------------------------------------------
// Kernel 1: 3x3 conv (implicit-GEMM) + bias + ReLU, fp32 WMMA 16x16x4.
// Block = 256 threads = 8 waves arranged 2(M) x 4(N); block tile M32 x N64.
// Double-buffered LDS pipeline with async global->LDS staging:
//   issue(s+1 -> buf^1); s_wait_asynccnt<=6; barrier; 4x WMMA on buf; barrier.
// Every lane always issues its async loads (full EXEC) so ASYNCcnt counts are
// exact; padded im2col lanes read from a zero-filled scratch line instead.
// ---------------------------------------------------------------------------
__global__ void __launch_bounds__(256)
conv3x3_wmma(const float* __restrict__ x, const float* __restrict__ w,
             const float* __restrict__ bias, const float* __restrict__ zpad,
             float* __restrict__ feat)
{
    __shared__ float lA[2][32 * LAPITCH];   // [m][c], padded pitch
    __shared__ float lB[2][KCH * 64];       // [c][n]

    const int tid  = threadIdx.x;
    const int lane = tid & 31;
    const int wave = tid >> 5;
    const int mt   = wave >> 2;     // 0..1
    const int nt   = wave & 3;      // 0..3
    const int hi   = lane >> 4;     // 0..1  (K half select)
    const int lo   = lane & 15;

    const int m0 = blockIdx.y * 32;     // gridDim.y = 1024
    const int n0 = blockIdx.x * 64;     // gridDim.x = 8

    // A-staging role: thread copies elements (am, ac) and (am, ac+8)
    const int am = tid >> 3;            // 0..31
    const int ac = tid & 7;             // 0..7
    const int gm = m0 + am;
    const int bi = gm >> 12;            // / 4096
    const int hw = gm & 4095;
    const int hh = hw >> 6;
    const int ww = hw & 63;

    // stage s = tap*32 + chunk ; c0 = chunk*16
    auto issue = [&](int s, int buf) {
        const int tap = s >> 5;
        const int c0  = (s & 31) << 4;
        const int dy = tap / 3 - 1;
        const int dx = tap % 3 - 1;
        const int ih = hh + dy;
        const int iw = ww + dx;
        const bool inb = (ih >= 0) && (ih < HH) && (iw >= 0) && (iw < WW);
        const float* a0 = inb ? (x + ((size_t)bi * CCH + c0 + ac) * HWSZ + ih * WW + iw)
                              : zpad;
        const float* a1 = inb ? (a0 + 8 * HWSZ) : zpad;
        async_copy_b32(a0, &lA[buf][am * LAPITCH + ac]);
        async_copy_b32(a1, &lA[buf][am * LAPITCH + ac + 8]);
        #pragma unroll
        for (int r = 0; r < 4; ++r) {
            const int e  = tid + r * 256;          // 0..1023
            const int cc = e >> 6;                 // 0..15
            const int nn = e & 63;
            async_copy_b32(w + (size_t)(n0 + nn) * KTOT + (size_t)(c0 + cc) * 9 + tap,
                           &lB[buf][e]);
        }
    };

    v8f acc = {};
    issue(0, 0);
    for (int s = 0; s < NSTG; ++s) {
        const int buf = s & 1;
        if (s + 1 < NSTG) {
            issue(s + 1, buf ^ 1);
            WAIT_ASYNC_LE(6);       // 6 copies of stage s+1 in flight -> stage s done
        } else {
            WAIT_ASYNC_LE(0);
        }
        __syncthreads();
        #pragma unroll
        for (int kk = 0; kk < 4; ++kk) {
            const int kb = kk * 4 + 2 * hi;   // K offset per f32 WMMA A/B layout
            v2f a, b;
            a.x = lA[buf][(mt * 16 + lo) * LAPITCH + kb];
            a.y = lA[buf][(mt * 16 + lo) * LAPITCH + kb + 1];
            b.x = lB[buf][kb * 64 + nt * 16 + lo];
            b.y = lB[buf][(kb + 1) * 64 + nt * 16 + lo];
            acc = __builtin_amdgcn_wmma_f32_16x16x4_f32(
                    false, a, false, b, (short)0, acc, false, false);
        }
        __syncthreads();
    }

    // epilogue: bias + ReLU, feat stored [M][512] row-major (NHWC-like)
    const int gn = n0 + nt * 16 + lo;
    const float bv = bias[gn];
    #pragma unroll
    for (int v = 0; v < 8; ++v) {
        const int gmr = m0 + mt * 16 + hi * 8 + v;
        float r = acc[v] + bv;
        feat[(size_t)gmr * CCH + gn] = r > 0.0f ? r : 0.0f;
    }
}

// ---------------------------------------------------------------------------
// Kernel 2a: build combined head weights [K=512][N=48] and zero the pad line
// ---------------------------------------------------------------------------
__global__ void build_headw(const float* __restrict__ bbox_w,
                            const float* __restrict__ cls_w,
                            float* __restrict__ headW,
                            float* __restrict__ zpad)
{
    int idx = blockIdx.x * 256 + threadIdx.x;
    if (idx < 16) zpad[idx] = 0.0f;
    if (idx >= CCH * NHEAD) return;
    int k = idx / NHEAD;
    int n = idx % NHEAD;
    float v = 0.0f;
    if (n < 36)      v = bbox_w[(size_t)n * CCH + k];
    else if (n < 45) v = cls_w[(size_t)(n - 36) * CCH + k];
    headW[(size_t)k * NHEAD + n] = v;
}

// ---------------------------------------------------------------------------
// Kernel 2b: 1x1 heads as WMMA GEMM  [32768 x 512] x [512 x 48]
// ---------------------------------------------------------------------------
__global__ void __launch_bounds__(256)
head_gemm(const float* __restrict__ feat, const float* __restrict__ headW,
          const float* __restrict__ bbox_b, const float* __restrict__ cls_b,
          float* __restrict__ headOut)
{
    const int wid  = blockIdx.x * 8 + (threadIdx.x >> 5);
    const int mt   = wid / 3;
    const int nt   = wid % 3;
    const int lane = threadIdx.x & 31;
    const int hi   = lane >> 4;
    const int lo   = lane & 15;

    v8f acc = {};
    const float* arow = feat + (size_t)(mt * 16 + lo) * CCH;
    for (int k0 = 0; k0 < CCH; k0 += 4) {
        if ((k0 & 127) == 0) __builtin_prefetch(arow + k0 + 128, 0, 1);
        const int kb = k0 + 2 * hi;
        v2f a, b;
        a.x = arow[kb];
        a.y = arow[kb + 1];
        b.x = headW[(size_t)kb * NHEAD + nt * 16 + lo];
        b.y = headW[(size_t)(kb + 1) * NHEAD + nt * 16 + lo];
        acc = __builtin_amdgcn_wmma_f32_16x16x4_f32(
                false, a, false, b, (short)0, acc, false, false);
    }
    const int gn = nt * 16 + lo;
    const float bv = (gn < 36) ? bbox_b[gn] : ((gn < 45) ? cls_b[gn - 36] : 0.0f);
    #pragma unroll
    for (int v = 0; v < 8; ++v) {
        const int gmr = mt * 16 + hi * 8 + v;
        headOut[(size_t)gmr * NHEAD + gn] = acc[v] + bv;
    }
}

// ---------------------------------------------------------------------------
// Kernel 3: box decode + clip + min-size masking
// ---------------------------------------------------------------------------
__global__ void decode_kernel(const float* __restrict__ headOut,
                              const float* __restrict__ anchors,
                              float* __restrict__ boxes,
                              float* __restrict__ scores)
{
    int idx = blockIdx.x * 256 + threadIdx.x;
    if (idx >= MTOT * ANC) return;
    const int a  = idx % ANC;
    const int m  = idx / ANC;           // b*4096 + h*64 + w
    const int hw = m & 4095;
    const int b  = m >> 12;

    const float* anc = anchors + ((size_t)hw * ANC + a) * 4;
    const float* ho  = headOut + (size_t)m * NHEAD;
    const float ddx = ho[a * 4 + 0], ddy = ho[a * 4 + 1];
    const float ddw = ho[a * 4 + 2], ddh = ho[a * 4 + 3];
    const float logit = ho[36 + a];
    const float px = anc[0], py = anc[1], pw = anc[2], ph = anc[3];

    const float cx = ddx * pw + px;
    const float cy = ddy * ph + py;
    const float bw = expf(ddw) * pw;
    const float bh = expf(ddh) * ph;
    float x1 = cx - 0.5f * bw, y1 = cy - 0.5f * bh;
    float x2 = cx + 0.5f * bw, y2 = cy + 0.5f * bh;
    x1 = fminf(fmaxf(x1, 0.0f), 63.0f);
    y1 = fminf(fmaxf(y1, 0.0f), 63.0f);
    x2 = fminf(fmaxf(x2, 0.0f), 63.0f);
    y2 = fminf(fmaxf(y2, 0.0f), 63.0f);
    const float wsz = x2 - x1, hsz = y2 - y1;
    const float sc = (wsz >= 1.0f && hsz >= 1.0f) ? logit : -1e9f;

    const size_t off = (size_t)b * (HWSZ * ANC) + (size_t)hw * ANC + a;
    boxes[off * 4 + 0] = x1;
    boxes[off * 4 + 1] = y1;
    boxes[off * 4 + 2] = x2;
    boxes[off * 4 + 3] = y2;
    scores[off] = sc;
}

// ---------------------------------------------------------------------------
// Kernel 4: per-batch pre-NMS top-2000 (histogram select + bitonic sort 2048)
// ---------------------------------------------------------------------------
__device__ __forceinline__ unsigned fkey(float f) {
    unsigned u = __float_as_uint(f);
    return (u & 0x80000000u) ? ~u : (u | 0x80000000u);
}

__global__ void __launch_bounds__(1024)
topk_kernel(const float* __restrict__ scores, const float* __restrict__ boxes,
            float* __restrict__ tscore, int* __restrict__ tidx,
            float* __restrict__ tboxes)
{
    __shared__ unsigned hist[4096];
    __shared__ float skey[2048];
    __shared__ int   sidx[2048];
    __shared__ int   shT, shAbove, cntHigh, cntTie;

    const int b = blockIdx.x;
    const int tid = threadIdx.x;
    const int N = HWSZ * ANC;                 // 36864
    const float* sc = scores + (size_t)b * N;

    for (int t = tid; t < 4096; t += 1024) hist[t] = 0u;
    __syncthreads();
    for (int i = tid; i < N; i += 1024)
        atomicAdd(&hist[fkey(sc[i]) >> 20], 1u);
    __syncthreads();
    if (tid == 0) {
        int cum = 0, T = 0;
        for (int bin = 4095; bin >= 0; --bin) {
            int h = (int)hist[bin];
            if (cum + h >= PRE_NMS) { T = bin; break; }
            cum += h;
        }
        shT = T; shAbove = cum; cntHigh = 0; cntTie = 0;
    }
    __syncthreads();
    for (int t = tid; t < 2048; t += 1024) { skey[t] = -__builtin_inff(); sidx[t] = 0; }
    __syncthreads();
    const int T = shT, above = shAbove;
    for (int i = tid; i < N; i += 1024) {
        const float s = sc[i];
        if ((int)(fkey(s) >> 20) > T) {
            int p = atomicAdd(&cntHigh, 1);
            skey[p] = s; sidx[p] = i;
        }
    }
    __syncthreads();
    for (int i = tid; i < N; i += 1024) {
        const float s = sc[i];
        if ((int)(fkey(s) >> 20) == T) {
            int p = above + atomicAdd(&cntTie, 1);
            if (p < PRE_NMS) { skey[p] = s; sidx[p] = i; }
        }
    }
    __syncthreads();

    // bitonic sort, descending by score, ascending idx as tiebreak
    for (int k = 2; k <= 2048; k <<= 1) {
        for (int j = k >> 1; j > 0; j >>= 1) {
            for (int t = tid; t < 2048; t += 1024) {
                const int ixj = t ^ j;
                if (ixj > t) {
                    float sa = skey[t], sb = skey[ixj];
                    int   ia = sidx[t], ib = sidx[ixj];
                    bool after  = (sa < sb) || (sa == sb && ia > ib);
                    bool dirAsc = ((t & k) == 0);
                    bool doSwap = dirAsc ? after : !after;
                    if (doSwap) {
                        skey[t] = sb; skey[ixj] = sa;
                        sidx[t] = ib; sidx[ixj] = ia;
                    }
                }
            }
            __syncthreads();
        }
    }

    for (int t = tid; t < 2048; t += 1024) {
        tscore[(size_t)b * 2048 + t] = skey[t];
        tidx[(size_t)b * 2048 + t] = sidx[t];
        const float* bp = boxes + ((size_t)b * N + sidx[t]) * 4;
        float* tb = tboxes + ((size_t)b * 2048 + t) * 4;
        tb[0] = bp[0]; tb[1] = bp[1]; tb[2] = bp[2]; tb[3] = bp[3];
    }
}

// ---------------------------------------------------------------------------
// Kernel 5: per-batch greedy NMS + post-NMS top-1000 (in-order compaction)
// ---------------------------------------------------------------------------
__global__ void __launch_bounds__(1024)
nms_kernel(const float* __restrict__ tboxes, const float* __restrict__ tscore,
           float* __restrict__ out)
{
    __shared__ float bx[PRE_NMS * 4];
    __shared__ float area[PRE_NMS];
    __shared__ int   keep[PRE_NMS];

    const int b = blockIdx.x;
    const int tid = threadIdx.x;

    for (int t = tid; t < PRE_NMS; t += 1024) {
        const float* tb = tboxes + ((size_t)b * 2048 + t) * 4;
        const float x1 = tb[0], y1 = tb[1], x2 = tb[2], y2 = tb[3];
        bx[t * 4 + 0] = x1; bx[t * 4 + 1] = y1;
        bx[t * 4 + 2] = x2; bx[t * 4 + 3] = y2;
        area[t] = (x2 - x1) * (y2 - y1);
        keep[t] = 1;
    }
    __syncthreads();

    for (int i = 0; i < PRE_NMS - 1; ++i) {
        if (keep[i]) {
            const float ix1 = bx[i * 4], iy1 = bx[i * 4 + 1];
            const float ix2 = bx[i * 4 + 2], iy2 = bx[i * 4 + 3];
            const float ia = area[i];
            for (int j = i + 1 + tid; j < PRE_NMS; j += 1024) {
                if (!keep[j]) continue;
                const float xx1 = fmaxf(ix1, bx[j * 4]);
                const float yy1 = fmaxf(iy1, bx[j * 4 + 1]);
                const float xx2 = fminf(ix2, bx[j * 4 + 2]);
                const float yy2 = fminf(iy2, bx[j * 4 + 3]);
                const float inter = fmaxf(xx2 - xx1, 0.0f) * fmaxf(yy2 - yy1, 0.0f);
                const float iou = inter / (ia + area[j] - inter);
                if (iou > NMS_T) keep[j] = 0;
            }
        }
        __syncthreads();
    }

    if (tid == 0) {
        float* ob = out + (size_t)b * POST_NMS * 5;
        int cnt = 0;
        for (int j = 0; j < PRE_NMS && cnt < POST_NMS; ++j) {
            const float s = tscore[(size_t)b * 2048 + j];
            if (keep[j] && s > -1e9f) {
                ob[cnt * 5 + 0] = bx[j * 4 + 0];
                ob[cnt * 5 + 1] = bx[j * 4 + 1];
                ob[cnt * 5 + 2] = bx[j * 4 + 2];
                ob[cnt * 5 + 3] = bx[j * 4 + 3];
                ob[cnt * 5 + 4] = s;
                ++cnt;
            }
        }
        for (int j = 0; j < PRE_NMS && cnt < POST_NMS; ++j) {
            const float s = tscore[(size_t)b * 2048 + j];
            if (!(keep[j] && s > -1e9f)) {
                ob[cnt * 5 + 0] = bx[j * 4 + 0];
                ob[cnt * 5 + 1] = bx[j * 4 + 1];
                ob[cnt * 5 + 2] = bx[j * 4 + 2];
                ob[cnt * 5 + 3] = bx[j * 4 + 3];
                ob[cnt * 5 + 4] = -1e9f;
                ++cnt;
            }
        }
    }
}

// ---------------------------------------------------------------------------
extern "C" void kernel_launch(void* const* d_in, const int* in_sizes, int n_in,
                              void* d_out, int out_size, void* d_ws, size_t ws_size,
                              hipStream_t stream)
{
    const float* x       = (const float*)d_in[0];
    const float* conv_w  = (const float*)d_in[1];
    const float* conv_b  = (const float*)d_in[2];
    const float* bbox_w  = (const float*)d_in[3];
    const float* bbox_b  = (const float*)d_in[4];
    const float* cls_w   = (const float*)d_in[5];
    const float* cls_b   = (const float*)d_in[6];
    const float* anchors = (const float*)d_in[7];
    float* out = (float*)d_out;

    float* ws      = (float*)d_ws;
    float* feat    = ws;                                    // 32768*512
    float* headW   = feat + (size_t)MTOT * CCH;             // 512*48
    float* headOut = headW + (size_t)CCH * NHEAD;           // 32768*48
    float* boxes   = headOut + (size_t)MTOT * NHEAD;        // 8*36864*4
    float* scores  = boxes + (size_t)BATCH * HWSZ * ANC * 4;// 8*36864
    float* tscore  = scores + (size_t)BATCH * HWSZ * ANC;   // 8*2048
    int*   tidx    = (int*)(tscore + BATCH * 2048);         // 8*2048
    float* tboxes  = (float*)(tidx + BATCH * 2048);         // 8*2048*4
    float* zpad    = tboxes + (size_t)BATCH * 2048 * 4;     // 16 (zero line)

    build_headw<<<(CCH * NHEAD + 255) / 256, 256, 0, stream>>>(bbox_w, cls_w, headW, zpad);
    conv3x3_wmma<<<dim3(8, 1024), 256, 0, stream>>>(x, conv_w, conv_b, zpad, feat);
    head_gemm<<<(2048 * 3) / 8, 256, 0, stream>>>(feat, headW, bbox_b, cls_b, headOut);
    decode_kernel<<<(MTOT * ANC + 255) / 256, 256, 0, stream>>>(headOut, anchors, boxes, scores);
    topk_kernel<<<BATCH, 1024, 0, stream>>>(scores, boxes, tscore, tidx, tboxes);
    nms_kernel<<<BATCH, 1024, 0, stream>>>(tboxes, tscore, out);
}